// GraphSAGE_31112743092745
// MI455X (gfx1250) — compile-verified
//
#include <hip/hip_runtime.h>
#include <hip/hip_bf16.h>
#include <math.h>

#define N_NODES   100000
#define N_EDGES   1600000
#define D_IN      128
#define D_HID     16
#define N_CLASSES 40

typedef float v2f __attribute__((ext_vector_type(2)));
typedef float v8f __attribute__((ext_vector_type(8)));

// ---------------------------------------------------------------------------
// GEMM1 (fused): Yl = X @ Wl, Yr = X @ Wr  computed in one pass over X.
// One wave owns one 16-row tile and BOTH 16-col outputs (two f32 accumulators);
// A-fragments are shared, so X (51 MB) is read exactly once.
// K swept in steps of 4 via V_WMMA_F32_16X16X4_F32; weights staged in LDS.
// ---------------------------------------------------------------------------
__global__ __launch_bounds__(128) void sage_gemm1_fused(
    const float* __restrict__ x,    // [N_NODES, 128] row-major
    const float* __restrict__ Wl,   // [128, 16] row-major
    const float* __restrict__ Wr,   // [128, 16] row-major
    float* __restrict__ Yl,         // [N_NODES, 16]
    float* __restrict__ Yr)         // [N_NODES, 16]
{
    __shared__ float sWl[D_IN * D_HID];   // 8 KB
    __shared__ float sWr[D_IN * D_HID];   // 8 KB
    for (int i = threadIdx.x; i < D_IN * D_HID; i += 128) {
        sWl[i] = Wl[i];
        sWr[i] = Wr[i];
    }
    __syncthreads();

    const int wave = threadIdx.x >> 5;
    const int lane = threadIdx.x & 31;
    const int tile = blockIdx.x * 4 + wave;
    if (tile >= N_NODES / 16) return;           // whole wave exits together
    const int row0 = tile * 16;

    const int half = lane >> 4;                 // 0: K pair {0,1}, 1: {2,3}
    const int r    = lane & 15;

    const float* xrow = x + (size_t)(row0 + r) * D_IN + 2 * half;

    v8f cl = {};
    v8f cr = {};
#pragma unroll
    for (int k0 = 0; k0 < D_IN; k0 += 4) {
        v2f a = *(const v2f*)(xrow + k0);       // 8B aligned (even element offset)
        const int kb = (k0 + 2 * half) * D_HID + r;
        v2f bl, br;
        bl.x = sWl[kb];  bl.y = sWl[kb + D_HID];
        br.x = sWr[kb];  br.y = sWr[kb + D_HID];
        cl = __builtin_amdgcn_wmma_f32_16x16x4_f32(
                false, a, false, bl, (short)0, cl, false, false);
        cr = __builtin_amdgcn_wmma_f32_16x16x4_f32(
                false, a, false, br, (short)0, cr, false, false);
    }

    // D layout: VGPR v -> M = v + 8*half, N = r
    const int mbase = row0 + half * 8;
#pragma unroll
    for (int v = 0; v < 8; ++v) {
        Yl[(size_t)(mbase + v) * D_HID + r] = cl[v];
        Yr[(size_t)(mbase + v) * D_HID + r] = cr[v];
    }
}

// ---------------------------------------------------------------------------
// Edge scatter-accumulate: sum[dst] += feat[src] (16 feats, 16 lanes/edge),
// optionally count in-degree once per edge. Gathers and atomics are 64B
// contiguous per edge; index reads hit one L0 line per 16-lane group.
// ---------------------------------------------------------------------------
__global__ __launch_bounds__(256) void sage_scatter_accum(
    const float* __restrict__ feat,        // [N_NODES, 16]
    const long long* __restrict__ src,     // [N_EDGES] (int64)
    const long long* __restrict__ dst,     // [N_EDGES]
    float* __restrict__ sum,               // [N_NODES, 16]
    float* __restrict__ cnt,               // [N_NODES] or nullptr
    int with_cnt)
{
    long long idx = (long long)blockIdx.x * 256 + threadIdx.x;
    if (idx >= (long long)N_EDGES * 16) return;
    const int e = (int)(idx >> 4);
    const int f = (int)(idx & 15);
    const int s = (int)src[e];
    const int d = (int)dst[e];
    atomicAdd(&sum[(size_t)d * 16 + f], feat[(size_t)s * 16 + f]);
    if (with_cnt && f == 0) atomicAdd(&cnt[d], 1.0f);
}

// h = relu(sum1 / max(cnt,1) + y1r + b1)
__global__ __launch_bounds__(256) void sage_combine_relu(
    const float* __restrict__ sum1, const float* __restrict__ cnt,
    const float* __restrict__ y1r,  const float* __restrict__ b1,
    float* __restrict__ h)
{
    int idx = blockIdx.x * 256 + threadIdx.x;
    if (idx >= N_NODES * 16) return;
    const int i = idx >> 4, f = idx & 15;
    const float cdeg = fmaxf(cnt[i], 1.0f);
    const float v = sum1[idx] / cdeg + y1r[idx] + b1[f];
    h[idx] = fmaxf(v, 0.0f);
}

// sum2 /= max(cnt,1)  (in place -> agg2)
__global__ __launch_bounds__(256) void sage_div_cnt(
    float* __restrict__ sum2, const float* __restrict__ cnt)
{
    int idx = blockIdx.x * 256 + threadIdx.x;
    if (idx >= N_NODES * 16) return;
    sum2[idx] /= fmaxf(cnt[idx >> 4], 1.0f);
}

// ---------------------------------------------------------------------------
// GEMM2: out[N,40] = agg2[N,16] @ W2l[16,40] + h[N,16] @ W2r[16,40] + b2
// Same accumulator receives both products; col tiles of 16 (40 -> 3 tiles).
// ---------------------------------------------------------------------------
__global__ __launch_bounds__(128) void sage_gemm2(
    const float* __restrict__ agg2, const float* __restrict__ h,
    const float* __restrict__ W2l,  const float* __restrict__ W2r,
    const float* __restrict__ b2,   float* __restrict__ out)
{
    const int wave = threadIdx.x >> 5;
    const int lane = threadIdx.x & 31;
    const int tile = blockIdx.x * 4 + wave;
    if (tile >= N_NODES / 16) return;
    const int row0 = tile * 16;
    const int c0   = blockIdx.y * 16;

    const int half = lane >> 4;
    const int r    = lane & 15;
    const int col  = c0 + r;
    const bool colok = (col < N_CLASSES);

    const float* arow = agg2 + (size_t)(row0 + r) * D_HID + 2 * half;
    const float* hrow = h    + (size_t)(row0 + r) * D_HID + 2 * half;

    v8f c = {};
#pragma unroll
    for (int k0 = 0; k0 < D_HID; k0 += 4) {
        v2f a = *(const v2f*)(arow + k0);
        const int k = k0 + 2 * half;
        v2f b;
        b.x = colok ? W2l[k * N_CLASSES + col]       : 0.0f;
        b.y = colok ? W2l[(k + 1) * N_CLASSES + col] : 0.0f;
        c = __builtin_amdgcn_wmma_f32_16x16x4_f32(
                false, a, false, b, (short)0, c, false, false);
    }
#pragma unroll
    for (int k0 = 0; k0 < D_HID; k0 += 4) {
        v2f a = *(const v2f*)(hrow + k0);
        const int k = k0 + 2 * half;
        v2f b;
        b.x = colok ? W2r[k * N_CLASSES + col]       : 0.0f;
        b.y = colok ? W2r[(k + 1) * N_CLASSES + col] : 0.0f;
        c = __builtin_amdgcn_wmma_f32_16x16x4_f32(
                false, a, false, b, (short)0, c, false, false);
    }

    if (colok) {
        const float bias = b2[col];
        const int mbase = row0 + half * 8;
#pragma unroll
        for (int v = 0; v < 8; ++v)
            out[(size_t)(mbase + v) * N_CLASSES + col] = c[v] + bias;
    }
}

// ---------------------------------------------------------------------------
// Row-wise log-softmax over 40 classes, in place. 256 rows per block staged
// through LDS: global traffic fully coalesced, LDS row stride padded to 41
// floats (gcd(41,64)=1) so the compute phase is bank-conflict free.
// ---------------------------------------------------------------------------
#define LSM_STRIDE 41
__global__ __launch_bounds__(256) void sage_log_softmax(float* __restrict__ out)
{
    __shared__ float srows[256 * LSM_STRIDE];   // ~41 KB
    const int row0 = blockIdx.x * 256;
    const int base = row0 * N_CLASSES;

    // coalesced load: 256*40 contiguous floats
    for (int i = threadIdx.x; i < 256 * N_CLASSES; i += 256) {
        const int g = base + i;
        const int rr = i / N_CLASSES, cc = i - rr * N_CLASSES;
        srows[rr * LSM_STRIDE + cc] = (g < N_NODES * N_CLASSES) ? out[g] : 0.0f;
    }
    __syncthreads();

    const int row = row0 + threadIdx.x;
    if (row < N_NODES) {
        float* p = &srows[threadIdx.x * LSM_STRIDE];
        float m = -INFINITY;
#pragma unroll
        for (int cidx = 0; cidx < N_CLASSES; ++cidx) m = fmaxf(m, p[cidx]);
        float s = 0.0f;
#pragma unroll
        for (int cidx = 0; cidx < N_CLASSES; ++cidx) s += expf(p[cidx] - m);
        const float ls = m + logf(s);
#pragma unroll
        for (int cidx = 0; cidx < N_CLASSES; ++cidx) p[cidx] = p[cidx] - ls;
    }
    __syncthreads();

    // coalesced store
    for (int i = threadIdx.x; i < 256 * N_CLASSES; i += 256) {
        const int g = base + i;
        const int rr = i / N_CLASSES, cc = i - rr * N_CLASSES;
        if (g < N_NODES * N_CLASSES) out[g] = srows[rr * LSM_STRIDE + cc];
    }
}

// ---------------------------------------------------------------------------
extern "C" void kernel_launch(void* const* d_in, const int* in_sizes, int n_in,
                              void* d_out, int out_size, void* d_ws, size_t ws_size,
                              hipStream_t stream)
{
    const float*     x   = (const float*)d_in[0];
    const long long* ei  = (const long long*)d_in[1];   // int64 [2, N_EDGES]
    const float*     W1l = (const float*)d_in[2];
    const float*     b1  = (const float*)d_in[3];
    const float*     W1r = (const float*)d_in[4];
    const float*     W2l = (const float*)d_in[5];
    const float*     b2  = (const float*)d_in[6];
    const float*     W2r = (const float*)d_in[7];
    float*           out = (float*)d_out;

    // Workspace layout (floats): y1l, y1r, sum1, cnt, h, sum2  (~32.4 MB)
    float* ws   = (float*)d_ws;
    float* y1l  = ws;
    float* y1r  = y1l  + (size_t)N_NODES * D_HID;
    float* sum1 = y1r  + (size_t)N_NODES * D_HID;
    float* cnt  = sum1 + (size_t)N_NODES * D_HID;
    float* hbuf = cnt  + (size_t)N_NODES;
    float* sum2 = hbuf + (size_t)N_NODES * D_HID;

    hipMemsetAsync(sum1, 0, (size_t)N_NODES * D_HID * sizeof(float), stream);
    hipMemsetAsync(cnt,  0, (size_t)N_NODES * sizeof(float), stream);
    hipMemsetAsync(sum2, 0, (size_t)N_NODES * D_HID * sizeof(float), stream);

    const long long* src  = ei;
    const long long* dstp = ei + N_EDGES;

    const int tiles  = N_NODES / 16;               // 6250
    const int g_gemm = (tiles + 3) / 4;            // 4 waves/block
    const long long etot = (long long)N_EDGES * 16;
    const int g_edge = (int)((etot + 255) / 256);
    const int g_node = (N_NODES * 16 + 255) / 256;

    // Layer 1: transform first (128 -> 16 twice, fused: X read once),
    // then aggregate 16-wide features.
    sage_gemm1_fused<<<g_gemm, 128, 0, stream>>>(x, W1l, W1r, y1l, y1r);
    sage_scatter_accum<<<g_edge, 256, 0, stream>>>(y1l, src, dstp, sum1, cnt, 1);
    sage_combine_relu<<<g_node, 256, 0, stream>>>(sum1, cnt, y1r, b1, hbuf);

    // Layer 2: aggregate h (16-wide), then transform 16 -> 40
    sage_scatter_accum<<<g_edge, 256, 0, stream>>>(hbuf, src, dstp, sum2, nullptr, 0);
    sage_div_cnt<<<g_node, 256, 0, stream>>>(sum2, cnt);
    sage_gemm2<<<dim3(g_gemm, 3), 128, 0, stream>>>(sum2, hbuf, W2l, W2r, b2, out);

    sage_log_softmax<<<(N_NODES + 255) / 256, 256, 0, stream>>>(out);
}